// Bilinear_48232482734312
// MI455X (gfx1250) — compile-verified
//
#include <hip/hip_runtime.h>
#include <hip/hip_bf16.h>

#define HQ 224
#define WQ 224

#ifndef __has_builtin
#define __has_builtin(x) 0
#endif

#if __has_builtin(__builtin_amdgcn_global_load_async_to_lds_b128)
#define HAVE_ASYNC_LD 1
#else
#define HAVE_ASYNC_LD 0
#endif

#if __has_builtin(__builtin_amdgcn_global_store_async_from_lds_b128)
#define HAVE_ASYNC_ST 1
#else
#define HAVE_ASYNC_ST 0
#endif

#if defined(__HIP_DEVICE_COMPILE__)
typedef int v4i __attribute__((ext_vector_type(4)));
typedef __attribute__((address_space(1))) v4i* gv4i_p;  // global (prints as __device__)
typedef __attribute__((address_space(3))) v4i* lv4i_p;  // LDS    (prints as __shared__)
#endif

__device__ __forceinline__ void wait_asynccnt0() {
#if __has_builtin(__builtin_amdgcn_s_wait_asynccnt)
    __builtin_amdgcn_s_wait_asynccnt(0);
#elif defined(__gfx1250__)
    asm volatile("s_wait_asynccnt 0" ::: "memory");
#endif
}

// One block = 256 threads = 256 consecutive pixels (flattened over B*H*W).
// Stage 1: async-DMA the block's contiguous 5120B input tile (256 px * 20B) to LDS.
// Stage 2: per-thread bilinear gather (4 x 3 global b32 loads, L2-resident).
// Stage 3: pack 3 f32/px into LDS, async-store 192 aligned b128 chunks to global.
__global__ __launch_bounds__(256) void Bilinear_48232482734312_kernel(
    const float* __restrict__ xin, float* __restrict__ outp, int npix) {
    __shared__ float lds_in[256 * 5];   // 5120 B, 16B-aligned
    __shared__ float lds_out[256 * 3];  // 3072 B, starts at 5120 -> 16B-aligned

    const int  t            = threadIdx.x;
    const long blockPixBase = (long)blockIdx.x * 256;
    const long inFloatBase  = blockPixBase * 5;

    // ---- stage input tile into LDS ----
    long availIn = (long)npix * 5 - inFloatBase;
    if (availIn > 1280) availIn = 1280;
    if (availIn < 0) availIn = 0;
    const int fullChunksIn = (int)(availIn >> 2);  // 16B chunks (=320 typical)
#if HAVE_ASYNC_LD
    for (int c = t; c < fullChunksIn; c += 256) {
        __builtin_amdgcn_global_load_async_to_lds_b128(
            (gv4i_p)(xin + inFloatBase + (long)c * 4),
            (lv4i_p)&lds_in[c * 4],
            0, 0);
    }
#else
    for (int c = t; c < fullChunksIn; c += 256) {
        const float4 v = *(const float4*)(xin + inFloatBase + (long)c * 4);
        *(float4*)&lds_in[c * 4] = v;
    }
#endif
    const int tailIn = (int)(availIn & 3);
    for (int i = t; i < tailIn; i += 256)
        lds_in[fullChunksIn * 4 + i] = xin[inFloatBase + fullChunksIn * 4 + i];
    wait_asynccnt0();
    __syncthreads();

    // ---- bilinear gather + blend ----
    const long pix = blockPixBase + t;
    float r = 0.f, g = 0.f, b = 0.f;
    if (pix < npix) {
        const float Xc = lds_in[t * 5 + 3];
        const float Yc = lds_in[t * 5 + 4];
        const long  bimg = pix / (HQ * WQ);
        const float* img0 = xin + bimg * (long)(HQ * WQ) * 5;

        const float fx = floorf(Xc), fy = floorf(Yc);
        const float wx = Xc - fx,    wy = Yc - fy;

        // padded-image indices, clipped to [0, W] / [0, H] like the reference
        const int px = (int)fminf(fmaxf(fx + 1.f, 0.f), (float)WQ);
        const int qx = (int)fminf(fmaxf(fx + 2.f, 0.f), (float)WQ);
        const int py = (int)fminf(fmaxf(fy + 1.f, 0.f), (float)HQ);
        const int qy = (int)fminf(fmaxf(fy + 2.f, 0.f), (float)HQ);

        // padded col/row 0 is the zero border; fold validity into the weights
        const float mx0 = (px >= 1) ? 1.f : 0.f;
        const float mx1 = (qx >= 1) ? 1.f : 0.f;
        const float my0 = (py >= 1) ? 1.f : 0.f;
        const float my1 = (qy >= 1) ? 1.f : 0.f;
        const int cx0 = (px >= 1) ? px - 1 : 0;
        const int cx1 = (qx >= 1) ? qx - 1 : 0;
        const int cy0 = (py >= 1) ? py - 1 : 0;
        const int cy1 = (qy >= 1) ? qy - 1 : 0;

        const float wtl = (1.f - wx) * (1.f - wy) * mx0 * my0;
        const float wtr = wx * (1.f - wy) * mx1 * my0;
        const float wbl = (1.f - wx) * wy * mx0 * my1;
        const float wbr = wx * wy * mx1 * my1;

        const float* ptl = img0 + ((long)cy0 * WQ + cx0) * 5;
        const float* ptr_ = img0 + ((long)cy0 * WQ + cx1) * 5;
        const float* pbl = img0 + ((long)cy1 * WQ + cx0) * 5;
        const float* pbr = img0 + ((long)cy1 * WQ + cx1) * 5;

        r = wtl * ptl[0] + wtr * ptr_[0] + wbl * pbl[0] + wbr * pbr[0];
        g = wtl * ptl[1] + wtr * ptr_[1] + wbl * pbl[1] + wbr * pbr[1];
        b = wtl * ptl[2] + wtr * ptr_[2] + wbl * pbl[2] + wbr * pbr[2];
    }
    lds_out[t * 3 + 0] = r;  // stride-3 across 64 banks: conflict-free
    lds_out[t * 3 + 1] = g;
    lds_out[t * 3 + 2] = b;
    __syncthreads();

    // ---- coalesced b128 writeback from LDS ----
    const long outFloatBase = blockPixBase * 3;  // *12B => 16B-aligned per block
    long availOut = (long)npix * 3 - outFloatBase;
    if (availOut > 768) availOut = 768;
    if (availOut < 0) availOut = 0;
    const int fullChunksOut = (int)(availOut >> 2);  // =192 typical
#if HAVE_ASYNC_ST
    for (int c = t; c < fullChunksOut; c += 256) {
        __builtin_amdgcn_global_store_async_from_lds_b128(
            (gv4i_p)(outp + outFloatBase + (long)c * 4),
            (lv4i_p)&lds_out[c * 4],
            0, 0);
    }
    wait_asynccnt0();
#else
    for (int c = t; c < fullChunksOut; c += 256) {
        const float4 v = *(const float4*)&lds_out[c * 4];
        *(float4*)(outp + outFloatBase + (long)c * 4) = v;
    }
#endif
    const int tailOut = (int)(availOut & 3);
    for (int i = t; i < tailOut; i += 256)
        outp[outFloatBase + fullChunksOut * 4 + i] = lds_out[fullChunksOut * 4 + i];
}

extern "C" void kernel_launch(void* const* d_in, const int* in_sizes, int n_in,
                              void* d_out, int out_size, void* d_ws, size_t ws_size,
                              hipStream_t stream) {
    const float* x   = (const float*)d_in[0];
    float*       out = (float*)d_out;
    const int npix   = in_sizes[0] / 5;           // 128*224*224 = 6,422,528
    const int blocks = (npix + 255) / 256;        // 25,088
    Bilinear_48232482734312_kernel<<<dim3(blocks), dim3(256), 0, stream>>>(x, out, npix);
}